// DirectionalAlignment_37761352466951
// MI455X (gfx1250) — compile-verified
//
#include <hip/hip_runtime.h>

typedef __attribute__((ext_vector_type(2))) float v2f;
typedef __attribute__((ext_vector_type(4))) float f4;
typedef __attribute__((ext_vector_type(8))) float v8f;
typedef __attribute__((ext_vector_type(4))) unsigned v4u;
typedef __attribute__((ext_vector_type(8))) int v8i;
typedef __attribute__((ext_vector_type(4))) int v4i;

#define S_DIM 256
#define D_DIM 256
#define P_DIM 64
#define B_DIM 8
#define LDS_STRIDE 260                      // 256 + 4 pad floats (TDM pad_amount=4 DWORDs)
#define LDS_X_FLOATS (S_DIM * LDS_STRIDE)   // 66560
#define LDS_RED_OFF  LDS_X_FLOATS           // 256 floats: red[s]
#define LDS_WRED_OFF (LDS_X_FLOATS + S_DIM) // 16 floats: per-wave partial sums
#define LDS_TOTAL_BYTES ((LDS_X_FLOATS + S_DIM + 16) * 4)  // 267,328 B < 320 KB

// ---------------------------------------------------------------------------
// K1: per (b,p) block.
//   - TDM tensor_load_to_lds stages X (256x256 f32) into LDS, hardware-padding
//     each 256-DWORD row with 4 DWORDs -> conflict-free 260-float row stride.
//   - scores = X X^T via V_WMMA_F32_16X16X4_F32, folded with W into red[s].
//   - y = x + red/sqrt(D) -> d_out, deterministic block partials -> ws.
// ---------------------------------------------------------------------------
extern "C" __global__ __launch_bounds__(256) void da_fused_kernel(
    const float* __restrict__ x, const float* __restrict__ w,
    float* __restrict__ y_out, float* __restrict__ partials)
{
  extern __shared__ float lds[];

  const int tid  = threadIdx.x;
  const int lane = tid & 31;
  const int wave = tid >> 5;
  const int half = lane >> 4;   // which 16-lane half of the wave
  const int m    = lane & 15;   // M (A) / N (B,C) coordinate

  const int bid = blockIdx.x;   // bid = b*P + p
  const int p   = bid & (P_DIM - 1);
  const float* __restrict__ xblk = x + (size_t)bid * (S_DIM * D_DIM);
  const float* __restrict__ wblk = w + (size_t)p * (S_DIM * S_DIM);

  // ---- Stage X via the Tensor Data Mover (wave 0 issues, all waves consume)
  if (wave == 0) {
    const unsigned long long gaddr = (unsigned long long)(const void*)xblk;
    const unsigned lds_off = (unsigned)(unsigned long long)(const void*)lds;

    // D# group 0: count=1 (valid), lds_addr, 57-bit global addr, type=2
    v4u g0;
    g0[0] = 1u;                                   // count=1, no gather
    g0[1] = lds_off;                              // LDS byte address
    g0[2] = (unsigned)(gaddr & 0xFFFFFFFFull);    // global_addr[31:0]
    g0[3] = (unsigned)((gaddr >> 32) & 0x1FFFFFFull) | (2u << 30); // [56:32] | type=2

    // D# group 1: mask=0, data_size=4B(2), pad_enable, pad_interval=256 DW(7),
    // pad_amount=4 DW(3); tensor 256x256, tile 256x256, dim0 stride 256.
    v8i g1;
    g1[0] = (int)((2u << 16) | (1u << 20) | (7u << 22) | (3u << 25));
    g1[1] = (int)(256u << 16);        // atomic_barrier_addr=0 | tensor_dim0[15:0]=256
    g1[2] = (int)(256u << 16);        // tensor_dim0[31:16]=0 | tensor_dim1[15:0]=256
    g1[3] = (int)(256u << 16);        // tensor_dim1[31:16]=0 | tile_dim0=256
    g1[4] = 256;                      // tile_dim1=256 | tile_dim2=0
    g1[5] = 256;                      // tensor_dim0_stride[31:0]=256
    g1[6] = 0;                        // stride0[47:32]=0 | dim1_stride lo=0
    g1[7] = 0;

    v4i g2 = {0, 0, 0, 0};            // 2D tensor: groups 2/3 unused (zero)
    v4i g3 = {0, 0, 0, 0};
    v8i g4 = {0, 0, 0, 0, 0, 0, 0, 0};

    __builtin_amdgcn_tensor_load_to_lds(g0, g1, g2, g3, g4, 0);
    __builtin_amdgcn_s_wait_tensorcnt(0);
  }
  __syncthreads();

  const float inv_dim = 0.0625f;  // 1/sqrt(256)

  // Each wave owns row-strips I0 = wave, I1 = wave+8; both K-swept together so
  // the 4 B-fragments are shared by 2 A-fragments: 6 ds loads -> 8 WMMAs.
  const int I0 = wave;
  const int I1 = wave + 8;
  const int a0row = I0 * 16 + m;
  const int a1row = I1 * 16 + m;

  float racc0[8], racc1[8];
#pragma unroll
  for (int r = 0; r < 8; ++r) { racc0[r] = 0.0f; racc1[r] = 0.0f; }

  // Keep exactly one J-group's accumulators live (stay under 256 VGPRs; no
  // s_set_vgpr_msb churn, 2 waves/SIMD fit alongside the 267 KB workgroup).
#pragma unroll 1
  for (int Jg = 0; Jg < 4; ++Jg) {
    const int j0 = Jg * 64;

    // Pull the W values the fold will need into cache while WMMAs run.
    __builtin_prefetch(wblk + (size_t)(I0 * 16 + m) * S_DIM + j0 + half * 32, 0, 0);
    __builtin_prefetch(wblk + (size_t)(I1 * 16 + m) * S_DIM + j0 + half * 32, 0, 0);

    v8f acc0[4], acc1[4];
#pragma unroll
    for (int j = 0; j < 4; ++j) { acc0[j] = (v8f){}; acc1[j] = (v8f){}; }

#pragma unroll 4
    for (int ks = 0; ks < 64; ++ks) {
      const int k = ks * 4 + 2 * half;
      const v2f a0 = *(const v2f*)&lds[a0row * LDS_STRIDE + k];
      const v2f a1 = *(const v2f*)&lds[a1row * LDS_STRIDE + k];
      const v2f b0 = *(const v2f*)&lds[(j0 +      m) * LDS_STRIDE + k];
      const v2f b1 = *(const v2f*)&lds[(j0 + 16 + m) * LDS_STRIDE + k];
      const v2f b2 = *(const v2f*)&lds[(j0 + 32 + m) * LDS_STRIDE + k];
      const v2f b3 = *(const v2f*)&lds[(j0 + 48 + m) * LDS_STRIDE + k];
      acc0[0] = __builtin_amdgcn_wmma_f32_16x16x4_f32(false, a0, false, b0,
                                                      (short)0, acc0[0], false, false);
      acc0[1] = __builtin_amdgcn_wmma_f32_16x16x4_f32(false, a0, false, b1,
                                                      (short)0, acc0[1], false, false);
      acc0[2] = __builtin_amdgcn_wmma_f32_16x16x4_f32(false, a0, false, b2,
                                                      (short)0, acc0[2], false, false);
      acc0[3] = __builtin_amdgcn_wmma_f32_16x16x4_f32(false, a0, false, b3,
                                                      (short)0, acc0[3], false, false);
      acc1[0] = __builtin_amdgcn_wmma_f32_16x16x4_f32(false, a1, false, b0,
                                                      (short)0, acc1[0], false, false);
      acc1[1] = __builtin_amdgcn_wmma_f32_16x16x4_f32(false, a1, false, b1,
                                                      (short)0, acc1[1], false, false);
      acc1[2] = __builtin_amdgcn_wmma_f32_16x16x4_f32(false, a1, false, b2,
                                                      (short)0, acc1[2], false, false);
      acc1[3] = __builtin_amdgcn_wmma_f32_16x16x4_f32(false, a1, false, b3,
                                                      (short)0, acc1[3], false, false);
    }

    // Fold finished score tiles with W immediately (scores never leave VGPRs).
    // C layout: VGPR r holds row M = I*16 + r + 8*half, col N = J*16 + m.
#pragma unroll
    for (int j = 0; j < 4; ++j) {
      const float* __restrict__ wr0 =
          wblk + (size_t)(I0 * 16) * S_DIM + (Jg * 4 + j) * 16 + m;
      const float* __restrict__ wr1 =
          wblk + (size_t)(I1 * 16) * S_DIM + (Jg * 4 + j) * 16 + m;
#pragma unroll
      for (int r = 0; r < 8; ++r) {
        racc0[r] += acc0[j][r] * wr0[(r + 8 * half) * S_DIM];
        racc1[r] += acc1[j][r] * wr1[(r + 8 * half) * S_DIM];
      }
    }
  }

  // Reduce racc over the 16 lanes of each half-wave -> red[s].
#pragma unroll
  for (int r = 0; r < 8; ++r) {
    float v0 = racc0[r], v1 = racc1[r];
    v0 += __shfl_xor(v0, 1, 32);  v1 += __shfl_xor(v1, 1, 32);
    v0 += __shfl_xor(v0, 2, 32);  v1 += __shfl_xor(v1, 2, 32);
    v0 += __shfl_xor(v0, 4, 32);  v1 += __shfl_xor(v1, 4, 32);
    v0 += __shfl_xor(v0, 8, 32);  v1 += __shfl_xor(v1, 8, 32);
    if (m == 0) {
      lds[LDS_RED_OFF + I0 * 16 + r + 8 * half] = v0 * inv_dim;
      lds[LDS_RED_OFF + I1 * 16 + r + 8 * half] = v1 * inv_dim;
    }
  }
  __syncthreads();

  // ---- y = x + red (X re-read from LDS, zero extra HBM), BN partials ----
  float s1 = 0.0f, s2 = 0.0f;
  float* __restrict__ yblk = y_out + (size_t)bid * (S_DIM * D_DIM);
  for (int i = tid; i < S_DIM * (D_DIM / 4); i += 256) {
    const int row = i >> 6;
    const int c4  = i & 63;
    f4 v = *(const f4*)&lds[row * LDS_STRIDE + c4 * 4];
    const float rv = lds[LDS_RED_OFF + row];
    v.x += rv; v.y += rv; v.z += rv; v.w += rv;
    ((f4*)yblk)[row * 64 + c4] = v;
    s1 += v.x + v.y + v.z + v.w;
    s2 += v.x * v.x + v.y * v.y + v.z * v.z + v.w * v.w;
  }
  for (int off = 1; off < 32; off <<= 1) {
    s1 += __shfl_xor(s1, off, 32);
    s2 += __shfl_xor(s2, off, 32);
  }
  if (lane == 0) {
    lds[LDS_WRED_OFF + wave]     = s1;
    lds[LDS_WRED_OFF + 8 + wave] = s2;
  }
  __syncthreads();
  if (tid == 0) {
    float t1 = 0.0f, t2 = 0.0f;
    for (int wv = 0; wv < 8; ++wv) {
      t1 += lds[LDS_WRED_OFF + wv];
      t2 += lds[LDS_WRED_OFF + 8 + wv];
    }
    partials[bid * 2]     = t1;   // deterministic: no float atomics
    partials[bid * 2 + 1] = t2;
  }
}

// ---------------------------------------------------------------------------
// K2: per-channel BN stats -> scale/shift (64 channels, trivial)
// ---------------------------------------------------------------------------
extern "C" __global__ void da_stats_kernel(
    const float* __restrict__ partials, const float* __restrict__ gamma,
    const float* __restrict__ beta, float* __restrict__ scaleshift)
{
  const int p = threadIdx.x;
  if (p >= P_DIM) return;
  float s1 = 0.0f, s2 = 0.0f;
  for (int b = 0; b < B_DIM; ++b) {
    s1 += partials[(b * P_DIM + p) * 2];
    s2 += partials[(b * P_DIM + p) * 2 + 1];
  }
  const float invN = 1.0f / ((float)B_DIM * S_DIM * D_DIM);
  const float mean = s1 * invN;
  const float var  = s2 * invN - mean * mean;
  const float sc   = gamma[p] * rsqrtf(var + 1e-5f);
  scaleshift[p * 2]     = sc;
  scaleshift[p * 2 + 1] = beta[p] - mean * sc;
}

// ---------------------------------------------------------------------------
// K3: normalize y in place
// ---------------------------------------------------------------------------
extern "C" __global__ __launch_bounds__(256) void da_norm_kernel(
    float* __restrict__ y, const float* __restrict__ scaleshift)
{
  const int bid = blockIdx.x;
  const int p   = bid & (P_DIM - 1);
  const float sc = scaleshift[p * 2];
  const float sh = scaleshift[p * 2 + 1];
  f4* __restrict__ yb = (f4*)(y + (size_t)bid * (S_DIM * D_DIM));
  for (int i = threadIdx.x; i < S_DIM * D_DIM / 4; i += 256) {
    f4 v = yb[i];
    v.x = v.x * sc + sh;
    v.y = v.y * sc + sh;
    v.z = v.z * sc + sh;
    v.w = v.w * sc + sh;
    yb[i] = v;
  }
}

// ---------------------------------------------------------------------------
extern "C" void kernel_launch(void* const* d_in, const int* in_sizes, int n_in,
                              void* d_out, int out_size, void* d_ws, size_t ws_size,
                              hipStream_t stream) {
  (void)in_sizes; (void)n_in; (void)out_size; (void)ws_size;
  const float* x     = (const float*)d_in[0];
  const float* w     = (const float*)d_in[1];
  const float* gamma = (const float*)d_in[2];
  const float* beta  = (const float*)d_in[3];
  float* out = (float*)d_out;

  float* partials   = (float*)d_ws;       // 1024 floats: (b,p) -> {sum, sumsq}
  float* scaleshift = partials + 1024;    // 128 floats:  p -> {scale, shift}

  const int nblk = B_DIM * P_DIM;         // 512
  da_fused_kernel<<<nblk, 256, LDS_TOTAL_BYTES, stream>>>(x, w, out, partials);
  da_stats_kernel<<<1, 64, 0, stream>>>(partials, gamma, beta, scaleshift);
  da_norm_kernel<<<nblk, 256, 0, stream>>>(out, scaleshift);
}